// GNNLayer_5325759447706
// MI455X (gfx1250) — compile-verified
//
#include <hip/hip_runtime.h>
#include <hip/hip_bf16.h>

typedef __attribute__((ext_vector_type(2))) float v2f;
typedef __attribute__((ext_vector_type(8))) float v8f;

#define MIN_NORM 1e-15f
#define BALL_EPS 0.004f
#define MIN_C 1e-6f

// ---------------- wave32 helpers ----------------
__device__ __forceinline__ float wredsum(float v) {
    v += __shfl_xor(v, 16, 32);
    v += __shfl_xor(v, 8, 32);
    v += __shfl_xor(v, 4, 32);
    v += __shfl_xor(v, 2, 32);
    v += __shfl_xor(v, 1, 32);
    return v;
}

__device__ __forceinline__ float artanhf_clip(float x) {
    x = fminf(fmaxf(x, -1.0f + 1e-5f), 1.0f - 1e-5f);
    return 0.5f * (log1pf(x) - log1pf(-x));
}

// ---------------- zero workspace ----------------
__global__ void zero_kernel(float* __restrict__ p, long long n) {
    long long i = (long long)blockIdx.x * blockDim.x + threadIdx.x;
    if (i < n) p[i] = 0.0f;
}

// ---------------- relation linear precompute: R1 = rela@Wr, R2 = rela@Wqr + b ----
__global__ void rel_linear_kernel(const float* __restrict__ rela,
                                  const float* __restrict__ Wr,
                                  const float* __restrict__ Wqr,
                                  const float* __restrict__ bqr,
                                  float* __restrict__ R1,
                                  float* __restrict__ R2) {
    const int r = blockIdx.x;
    const int c = threadIdx.x;   // 0..63
    const float* row = rela + (size_t)r * 64;
    float s1 = 0.f, s2 = 0.f;
#pragma unroll
    for (int k = 0; k < 64; ++k) {
        float rv = row[k];
        s1 = fmaf(rv, Wr[k * 64 + c], s1);
        s2 = fmaf(rv, Wqr[k * 64 + c], s2);
    }
    R1[(size_t)r * 64 + c] = s1;
    R2[(size_t)r * 64 + c] = s2 + bqr[c];
}

// ---------------- expmap0+project precompute (wave per row) ----------------
// OUT[i] = project(expmap0(IN[i], c), c); N2[i] = ||OUT[i]||^2
__global__ void exp0_kernel(const float* __restrict__ in,
                            const float* __restrict__ curv,
                            float* __restrict__ out,
                            float* __restrict__ n2out,
                            int nrows) {
    const int lane = threadIdx.x & 31;
    const int wid  = (blockIdx.x * blockDim.x + threadIdx.x) >> 5;
    const int nw   = (gridDim.x * blockDim.x) >> 5;
    const float cc = fmaxf(curv[0], MIN_C);
    const float sc = sqrtf(cc);
    const float maxn = (1.0f - BALL_EPS) / sc;
    for (int i = wid; i < nrows; i += nw) {
        float2 u = *(const float2*)(in + (size_t)i * 64 + 2 * lane);
        float un = fmaxf(sqrtf(wredsum(u.x * u.x + u.y * u.y)), MIN_NORM);
        float t  = tanhf(fminf(fmaxf(sc * un, -15.f), 15.f));
        float g  = t / (sc * un);
        float gx = u.x * g, gy = u.y * g;
        float nn = fmaxf(sqrtf(wredsum(gx * gx + gy * gy)), MIN_NORM);
        float s  = (nn > maxn) ? (maxn / nn) : 1.0f;
        gx *= s; gy *= s;
        float2 o; o.x = gx; o.y = gy;
        *(float2*)(out + (size_t)i * 64 + 2 * lane) = o;
        if (lane == 0) n2out[i] = (nn * s) * (nn * s);
    }
}

// ---------------- fp32 WMMA GEMM: C(Mx64) = A(Mx64) @ W(64x64) --------------
// One wave per 16x16 output tile; V_WMMA_F32_16X16X4_F32, K swept in 16 steps.
__global__ void gemm64_wmma(const float* __restrict__ A,
                            const float* __restrict__ W,
                            float* __restrict__ C,
                            int M) {
    const int lane   = threadIdx.x & 31;
    const int tile   = blockIdx.x * (blockDim.x >> 5) + (threadIdx.x >> 5);
    const int ntiles = (M >> 4) * 4;          // M assumed multiple of 16 (200000 is)
    if (tile >= ntiles) return;               // wave-uniform guard (EXEC stays full)
    const int m0   = (tile >> 2) << 4;
    const int col0 = (tile & 3) << 4;
    const int mr   = m0 + (lane & 15);        // A row for this lane
    const int kk   = (lane >> 4) << 1;        // K sub-offset: 0 or 2
    const int nc   = col0 + (lane & 15);      // B/C column for this lane
    v8f acc = {};
#pragma unroll
    for (int k0 = 0; k0 < 64; k0 += 4) {
        v2f a, b;
        a.x = A[(size_t)mr * 64 + k0 + kk];
        a.y = A[(size_t)mr * 64 + k0 + kk + 1];
        b.x = W[(size_t)(k0 + kk) * 64 + nc];
        b.y = W[(size_t)(k0 + kk + 1) * 64 + nc];
        acc = __builtin_amdgcn_wmma_f32_16x16x4_f32(
            false, a, false, b, (short)0, acc, false, false);
    }
    const int rbase = m0 + ((lane >> 4) << 3);
#pragma unroll
    for (int v = 0; v < 8; ++v)
        C[(size_t)(rbase + v) * 64 + nc] = acc[v];
}

// ---------------- per-edge kernel: one wave per edge, 2 dims per lane --------
__global__ void edge_kernel(const float* __restrict__ EX, const float* __restrict__ X2,
                            const float* __restrict__ EY, const float* __restrict__ Y2,
                            const float* __restrict__ P,  const float* __restrict__ R1,
                            const float* __restrict__ R2,
                            const float* __restrict__ walpha_w,
                            const float* __restrict__ walpha_b,
                            const float* __restrict__ curv,
                            const int* __restrict__ sub_idx,
                            const int* __restrict__ rel_idx,
                            const int* __restrict__ obj_idx,
                            const int* __restrict__ qr_idx,
                            float* __restrict__ agg,
                            int n_edge) {
    const int lane = threadIdx.x & 31;
    const int wid  = (blockIdx.x * blockDim.x + threadIdx.x) >> 5;
    const int nw   = (gridDim.x * blockDim.x) >> 5;
    const float cc = fmaxf(curv[0], MIN_C);
    const float sc = sqrtf(cc);
    const float maxn = (1.0f - BALL_EPS) / sc;
    const float2 wa = *(const float2*)(walpha_w + 2 * lane);
    const float wab = walpha_b[0];

    for (int e = wid; e < n_edge; e += nw) {
        const int s = sub_idx[e], r = rel_idx[e], o = obj_idx[e], q = qr_idx[e];

        // prefetch next iteration's node rows into cache
        const int en = e + nw;
        if (en < n_edge) {
            const int sn = sub_idx[en];
            __builtin_prefetch(EX + (size_t)sn * 64, 0, 1);
            __builtin_prefetch(P  + (size_t)sn * 64, 0, 1);
        }

        const float2 x = *(const float2*)(EX + (size_t)s * 64 + 2 * lane);
        const float2 y = *(const float2*)(EY + (size_t)r * 64 + 2 * lane);
        const float x2 = X2[s];
        const float y2 = Y2[r];

        // mobius_add(x, y, c)
        const float xy  = wredsum(x.x * y.x + x.y * y.y);
        const float ca  = 1.f + 2.f * cc * xy + cc * y2;
        const float cb  = 1.f - cc * x2;
        const float den = fmaxf(1.f + 2.f * cc * xy + cc * cc * x2 * y2, MIN_NORM);
        float mx = (ca * x.x + cb * y.x) / den;
        float my = (ca * x.y + cb * y.y) / den;

        // project + logmap0
        const float mn = fmaxf(sqrtf(wredsum(mx * mx + my * my)), MIN_NORM);
        const float ps = (mn > maxn) ? (maxn / mn) : 1.0f;
        const float yn = fmaxf(mn * ps, MIN_NORM);
        const float ls = ps * (artanhf_clip(sc * yn) / (yn * sc));
        const float msgx = mx * ls, msgy = my * ls;

        // attention gate: sigmoid(relu(P[s]+R1[r]+R2[q]) . walpha + b)
        const float2 p  = *(const float2*)(P  + (size_t)s * 64 + 2 * lane);
        const float2 r1 = *(const float2*)(R1 + (size_t)r * 64 + 2 * lane);
        const float2 r2 = *(const float2*)(R2 + (size_t)q * 64 + 2 * lane);
        const float t0 = fmaxf(p.x + r1.x + r2.x, 0.f);
        const float t1 = fmaxf(p.y + r1.y + r2.y, 0.f);
        const float dot = wredsum(t0 * wa.x + t1 * wa.y);
        const float alpha = 1.0f / (1.0f + expf(-(dot + wab)));

        // scatter-add to destination node (hardware f32 atomics)
        unsafeAtomicAdd(agg + (size_t)o * 64 + 2 * lane,     alpha * msgx);
        unsafeAtomicAdd(agg + (size_t)o * 64 + 2 * lane + 1, alpha * msgy);
    }
}

extern "C" void kernel_launch(void* const* d_in, const int* in_sizes, int n_in,
                              void* d_out, int out_size, void* d_ws, size_t ws_size,
                              hipStream_t stream) {
    const float* hidden  = (const float*)d_in[0];
    const float* rela    = (const float*)d_in[1];
    const float* Ws_w    = (const float*)d_in[2];
    const float* Wr_w    = (const float*)d_in[3];
    const float* Wqr_w   = (const float*)d_in[4];
    const float* Wqr_b   = (const float*)d_in[5];
    const float* walphaw = (const float*)d_in[6];
    const float* walphab = (const float*)d_in[7];
    const float* Wh_w    = (const float*)d_in[8];
    const float* curv    = (const float*)d_in[9];
    const int*   sub_idx = (const int*)d_in[10];
    const int*   rel_idx = (const int*)d_in[11];
    const int*   obj_idx = (const int*)d_in[12];
    const int*   qr_idx  = (const int*)d_in[13];

    const int NN = in_sizes[0] / 64;      // 200000 nodes
    const int NR = in_sizes[1] / 64;      // 401 relations
    const int NE = in_sizes[10];          // 1e6 edges

    // workspace layout (floats)
    float* ws = (float*)d_ws;
    const size_t nd = (size_t)NN * 64;
    float* P   = ws;                      // hidden @ Ws
    float* EX  = ws + nd;                 // project(expmap0(hidden))
    float* agg = ws + 2 * nd;             // scatter accumulator
    float* X2  = ws + 3 * nd;             // ||EX||^2 per node
    float* R1  = X2 + NN;                 // rela @ Wr
    float* R2  = R1 + (size_t)NR * 64;    // rela @ Wqr + b
    float* EY  = R2 + (size_t)NR * 64;    // project(expmap0(rela))
    float* Y2  = EY + (size_t)NR * 64;    // ||EY||^2 per relation

    // 1) zero accumulator
    {
        long long n = (long long)nd;
        int blocks = (int)((n + 255) / 256);
        zero_kernel<<<blocks, 256, 0, stream>>>(agg, n);
    }
    // 2) relation tables
    rel_linear_kernel<<<NR, 64, 0, stream>>>(rela, Wr_w, Wqr_w, Wqr_b, R1, R2);
    exp0_kernel<<<64, 256, 0, stream>>>(rela, curv, EY, Y2, NR);
    // 3) node precompute
    exp0_kernel<<<4096, 256, 0, stream>>>(hidden, curv, EX, X2, NN);
    {   // P = hidden @ Ws  (fp32 WMMA)
        int tiles = (NN / 16) * 4;
        gemm64_wmma<<<(tiles + 7) / 8, 256, 0, stream>>>(hidden, Ws_w, P, NN);
    }
    // 4) per-edge message + attention + scatter
    edge_kernel<<<4096, 256, 0, stream>>>(EX, X2, EY, Y2, P, R1, R2,
                                          walphaw, walphab, curv,
                                          sub_idx, rel_idx, obj_idx, qr_idx,
                                          agg, NE);
    // 5) out = agg @ Wh  (fp32 WMMA)
    {
        int tiles = (NN / 16) * 4;
        gemm64_wmma<<<(tiles + 7) / 8, 256, 0, stream>>>(agg, Wh_w, (float*)d_out, NN);
    }
}